// GraphNN_83064667504669
// MI455X (gfx1250) — compile-verified
//
#include <hip/hip_runtime.h>
#include <hip/hip_bf16.h>

typedef float v2f __attribute__((ext_vector_type(2)));
typedef float v8f __attribute__((ext_vector_type(8)));

#define NHEADS 3

__device__ __forceinline__ float lrelu(float x, float s) { return x > 0.0f ? x : s * x; }

__device__ __forceinline__ float atomicMaxFloat(float* addr, float val) {
  unsigned int* ua = (unsigned int*)addr;
  unsigned int old = *ua;
  while (__uint_as_float(old) < val) {
    unsigned int assumed = old;
    old = atomicCAS(ua, assumed, __float_as_uint(val));
    if (old == assumed) break;
  }
  return __uint_as_float(old);
}

// ---------------------------------------------------------------------------
// One wave (32 threads) computes one 16x16 tile of C = A[MxK] @ B[KxNcols]
// with V_WMMA_F32_16X16X4_F32.  K is compile-time; GUARD selects between a
// guard-free fast path (tile fully interior; chosen by a wave-uniform scalar
// branch so EXEC stays all-1s) and a branchless-clamped edge path.
// A frag (16x4 f32): lane row=lane%16, v0=K(2*half), v1=K(2*half+1).
// C/D: lane col=lane%16, vgpr j holds M = j + 8*(lane/16).
// ---------------------------------------------------------------------------
template <int K, bool GUARD>
__device__ __forceinline__ void wmma_tile(const float* __restrict__ A,
                                          const float* __restrict__ B,
                                          float* __restrict__ C,
                                          int M, int Ncols, int m0, int n0,
                                          int hl, int lr) {
  const int row = m0 + lr;
  const int col = n0 + lr;
  bool rvalid = true, cvalid = true;
  int rrow = row, ccol = col;
  if (GUARD) {
    rvalid = row < M;      rrow = rvalid ? row : (M - 1);
    cvalid = col < Ncols;  ccol = cvalid ? col : 0;
  }

  constexpr int KSTEPS = (K + 3) / 4;
  v2f a[KSTEPS], b[KSTEPS];
#pragma unroll
  for (int s = 0; s < KSTEPS; ++s) {
    const int ka0 = s * 4 + 2 * hl;
    const int ka1 = ka0 + 1;
    float ax = (ka0 < K) ? A[(long)rrow * K + ka0] : 0.0f;
    float ay = (ka1 < K) ? A[(long)rrow * K + ka1] : 0.0f;
    float bx = (ka0 < K) ? B[(long)ka0 * Ncols + ccol] : 0.0f;
    float by = (ka1 < K) ? B[(long)ka1 * Ncols + ccol] : 0.0f;
    if (GUARD) {
      ax = rvalid ? ax : 0.0f;
      ay = rvalid ? ay : 0.0f;
      bx = cvalid ? bx : 0.0f;
      by = cvalid ? by : 0.0f;
    }
    a[s].x = ax; a[s].y = ay;
    b[s].x = bx; b[s].y = by;
  }

  v8f acc = {};
#pragma unroll
  for (int s = 0; s < KSTEPS; ++s)
    acc = __builtin_amdgcn_wmma_f32_16x16x4_f32(false, a[s], false, b[s], (short)0, acc,
                                                false, false);
  if (!GUARD || cvalid) {
#pragma unroll
    for (int j = 0; j < 8; ++j) {
      int m = m0 + j + 8 * hl;
      if (!GUARD || m < M) C[(long)m * Ncols + col] = acc[j];
    }
  }
}

template <int K>
__global__ void wmma_gemm_f32_k(const float* __restrict__ A, const float* __restrict__ B,
                                float* __restrict__ C, int M, int Ncols) {
  const int lane = threadIdx.x;  // 0..31
  const int hl   = lane >> 4;
  const int lr   = lane & 15;
  const int m0   = blockIdx.x * 16;
  const int n0   = blockIdx.y * 16;
  if (m0 + 16 <= M && n0 + 16 <= Ncols)
    wmma_tile<K, false>(A, B, C, M, Ncols, m0, n0, hl, lr);   // interior: no guards
  else
    wmma_tile<K, true>(A, B, C, M, Ncols, m0, n0, hl, lr);    // ragged tail tile
}

__global__ void fill_kernel(float* __restrict__ p, float v, long n) {
  long i = (long)blockIdx.x * blockDim.x + threadIdx.x;
  long strd = (long)gridDim.x * blockDim.x;
  for (; i < n; i += strd) p[i] = v;
}

// c[h] = sum_d We[0, h*D+d] * ae[h, d]   (fe = ef * We row, so ee = ef * c)
__global__ void compute_c_kernel(const float* __restrict__ We, const float* __restrict__ ae,
                                 float* __restrict__ c, int D) {
  int h = threadIdx.x;
  if (h < NHEADS) {
    float s = 0.0f;
    for (int d = 0; d < D; ++d) s += We[h * D + d] * ae[h * D + d];
    c[h] = s;
  }
}

// el[n,h] = ft[n,h,:].al[h,:]   er[n,h] = ft[n,h,:].ar[h,:]
__global__ void attn_node_kernel(const float* __restrict__ ft, const float* __restrict__ al,
                                 const float* __restrict__ ar, float* __restrict__ el,
                                 float* __restrict__ er, int N, int D) {
  int n = blockIdx.x * blockDim.x + threadIdx.x;
  if (n >= N) return;
  const float* f = ft + (long)n * NHEADS * D;
  for (int h = 0; h < NHEADS; ++h) {
    float sl = 0.0f, sr = 0.0f;
    for (int d = 0; d < D; ++d) {
      float v = f[h * D + d];
      sl += v * al[h * D + d];
      sr += v * ar[h * D + d];
    }
    el[n * NHEADS + h] = sl;
    er[n * NHEADS + h] = sr;
  }
}

// e = leakyrelu(el[src]+er[dst]+ef*c, 0.2); segment-max into m[dst]
__global__ void edge_logits_max_kernel(const float* __restrict__ el, const float* __restrict__ er,
                                       const float* __restrict__ ef, const float* __restrict__ c,
                                       const int* __restrict__ src, const int* __restrict__ dst,
                                       float* __restrict__ ev, float* __restrict__ mb, int E) {
  int e = blockIdx.x * blockDim.x + threadIdx.x;
  if (e >= E) return;
  int s = src[e], d2 = dst[e];
  float fe = ef[e];
  for (int h = 0; h < NHEADS; ++h) {
    float raw = el[s * NHEADS + h] + er[d2 * NHEADS + h] + fe * c[h];
    float v = lrelu(raw, 0.2f);
    ev[(long)e * NHEADS + h] = v;
    atomicMaxFloat(&mb[d2 * NHEADS + h], v);
  }
}

// ev <- exp(ev - m[dst]); segment-sum into s[dst]
__global__ void edge_exp_kernel(float* __restrict__ ev, const float* __restrict__ mb,
                                const int* __restrict__ dst, float* __restrict__ sb, int E) {
  int e = blockIdx.x * blockDim.x + threadIdx.x;
  if (e >= E) return;
  int d2 = dst[e];
  for (int h = 0; h < NHEADS; ++h) {
    float x = __expf(ev[(long)e * NHEADS + h] - mb[d2 * NHEADS + h]);
    ev[(long)e * NHEADS + h] = x;
    atomicAdd(&sb[d2 * NHEADS + h], x);
  }
}

// rst[dst] += ft[src] * (ev / s[dst]) ; one thread per (edge, head)
__global__ void aggregate_kernel(const float* __restrict__ ev, const float* __restrict__ sb,
                                 const float* __restrict__ ft, const int* __restrict__ src,
                                 const int* __restrict__ dst, float* __restrict__ rst,
                                 int E, int D) {
  long tid = (long)blockIdx.x * blockDim.x + threadIdx.x;
  if (tid >= (long)E * NHEADS) return;
  int e = (int)(tid / NHEADS);
  int h = (int)(tid % NHEADS);
  int s = src[e], d2 = dst[e];
  float a = ev[(long)e * NHEADS + h] / sb[d2 * NHEADS + h];
  const float* fsrc = ft + (long)s * NHEADS * D + h * D;
  float* rdst = rst + (long)d2 * NHEADS * D + h * D;
  for (int d = 0; d < D; ++d) atomicAdd(&rdst[d], fsrc[d] * a);
}

// rst <- rst + res + b (in place); hnext[n,d] = mean_h rst[n,h,d]
__global__ void finalize_kernel(float* __restrict__ rst, const float* __restrict__ res,
                                const float* __restrict__ b, float* __restrict__ hnext,
                                int N, int D) {
  long tid = (long)blockIdx.x * blockDim.x + threadIdx.x;
  if (tid >= (long)N * D) return;
  int n = (int)(tid / D);
  int d = (int)(tid % D);
  float sum = 0.0f;
  for (int h = 0; h < NHEADS; ++h) {
    long idx = (long)n * NHEADS * D + h * D + d;
    float v = rst[idx] + res[idx] + b[h * D + d];
    rst[idx] = v;
    sum += v;
  }
  hnext[(long)n * D + d] = sum * (1.0f / NHEADS);
}

// per-node MLP head + global max
__global__ void mlp_max_kernel(const float* __restrict__ h3, const float* __restrict__ L1a_w,
                               const float* __restrict__ L1a_b, const float* __restrict__ L1b_w,
                               const float* __restrict__ L1b_b, const float* __restrict__ L1c_w,
                               const float* __restrict__ L1c_b, const float* __restrict__ L2_w,
                               const float* __restrict__ L2_b, float* __restrict__ out, int N) {
  int n = blockIdx.x * blockDim.x + threadIdx.x;
  float z = -__builtin_inff();
  if (n < N) {
    float acc2 = L2_b[0];
    for (int h = 0; h < NHEADS; ++h) {
      const float* x = h3 + (long)n * NHEADS * 32 + h * 32;
      float y16[16];
      for (int j = 0; j < 16; ++j) {
        float s = L1a_b[j];
        for (int d = 0; d < 32; ++d) s += x[d] * L1a_w[d * 16 + j];
        y16[j] = lrelu(s, 0.01f);
      }
      float y4[4];
      for (int j = 0; j < 4; ++j) {
        float s = L1b_b[j];
        for (int d = 0; d < 16; ++d) s += y16[d] * L1b_w[d * 4 + j];
        y4[j] = lrelu(s, 0.01f);
      }
      float y1 = L1c_b[0];
      for (int d = 0; d < 4; ++d) y1 += y4[d] * L1c_w[d];
      acc2 += y1 * L2_w[h];
    }
    z = acc2;
  }
  __shared__ float red[256];
  red[threadIdx.x] = z;
  __syncthreads();
  for (int sft = blockDim.x / 2; sft > 0; sft >>= 1) {
    if ((int)threadIdx.x < sft) red[threadIdx.x] = fmaxf(red[threadIdx.x], red[threadIdx.x + sft]);
    __syncthreads();
  }
  if (threadIdx.x == 0) atomicMaxFloat(out, red[0]);
}

extern "C" void kernel_launch(void* const* d_in, const int* in_sizes, int n_in,
                              void* d_out, int out_size, void* d_ws, size_t ws_size,
                              hipStream_t stream) {
  (void)n_in; (void)out_size; (void)ws_size;
  const int N = in_sizes[0] / 6;       // node count
  const int E = in_sizes[31];          // edge count (src)
  const float* node_feats = (const float*)d_in[0];
  const float* edge_feats = (const float*)d_in[1];
  const int* src = (const int*)d_in[31];
  const int* dst = (const int*)d_in[32];

  // carve workspace (256B-aligned chunks)
  size_t off = 0;
  auto alloc = [&](size_t nfloats) {
    float* p = (float*)((char*)d_ws + off);
    off += ((nfloats * sizeof(float) + 255) / 256) * 256;
    return p;
  };
  float* ft    = alloc((size_t)N * 96);
  float* res   = alloc((size_t)N * 96);
  float* rst   = alloc((size_t)N * 96);
  float* hnext = alloc((size_t)N * 32);
  float* el    = alloc((size_t)N * NHEADS);
  float* er    = alloc((size_t)N * NHEADS);
  float* mb    = alloc((size_t)N * NHEADS);
  float* sb    = alloc((size_t)N * NHEADS);
  float* ev    = alloc((size_t)E * NHEADS);
  float* cbuf  = alloc(16);

  struct Layer { int din, D; const float *W, *We, *al, *ar, *ae, *Wr, *b; };
  Layer L[3] = {
    { 6,  8, (const float*)d_in[2],  (const float*)d_in[3],  (const float*)d_in[4],
             (const float*)d_in[5],  (const float*)d_in[6],  (const float*)d_in[7],
             (const float*)d_in[8] },
    { 8, 16, (const float*)d_in[9],  (const float*)d_in[10], (const float*)d_in[11],
             (const float*)d_in[12], (const float*)d_in[13], (const float*)d_in[14],
             (const float*)d_in[15] },
    {16, 32, (const float*)d_in[16], (const float*)d_in[17], (const float*)d_in[18],
             (const float*)d_in[19], (const float*)d_in[20], (const float*)d_in[21],
             (const float*)d_in[22] },
  };

  const float NEG_INF = -__builtin_inff();
  const float* hin = node_feats;
  for (int l = 0; l < 3; ++l) {
    const int din = L[l].din, D = L[l].D, dout = NHEADS * D;
    dim3 gG((N + 15) / 16, (dout + 15) / 16);
    switch (din) {
      case 6:
        wmma_gemm_f32_k<6><<<gG, 32, 0, stream>>>(hin, L[l].W,  ft,  N, dout);
        wmma_gemm_f32_k<6><<<gG, 32, 0, stream>>>(hin, L[l].Wr, res, N, dout);
        break;
      case 8:
        wmma_gemm_f32_k<8><<<gG, 32, 0, stream>>>(hin, L[l].W,  ft,  N, dout);
        wmma_gemm_f32_k<8><<<gG, 32, 0, stream>>>(hin, L[l].Wr, res, N, dout);
        break;
      default:
        wmma_gemm_f32_k<16><<<gG, 32, 0, stream>>>(hin, L[l].W,  ft,  N, dout);
        wmma_gemm_f32_k<16><<<gG, 32, 0, stream>>>(hin, L[l].Wr, res, N, dout);
        break;
    }
    compute_c_kernel<<<1, 32, 0, stream>>>(L[l].We, L[l].ae, cbuf, D);
    attn_node_kernel<<<(N + 255) / 256, 256, 0, stream>>>(ft, L[l].al, L[l].ar, el, er, N, D);
    fill_kernel<<<1024, 256, 0, stream>>>(mb, NEG_INF, (long)N * NHEADS);
    fill_kernel<<<1024, 256, 0, stream>>>(sb, 0.0f, (long)N * NHEADS);
    fill_kernel<<<4096, 256, 0, stream>>>(rst, 0.0f, (long)N * dout);
    edge_logits_max_kernel<<<(E + 255) / 256, 256, 0, stream>>>(el, er, edge_feats, cbuf,
                                                                src, dst, ev, mb, E);
    edge_exp_kernel<<<(E + 255) / 256, 256, 0, stream>>>(ev, mb, dst, sb, E);
    long agg_threads = (long)E * NHEADS;
    aggregate_kernel<<<(int)((agg_threads + 255) / 256), 256, 0, stream>>>(ev, sb, ft, src, dst,
                                                                           rst, E, D);
    long fin_threads = (long)N * D;
    finalize_kernel<<<(int)((fin_threads + 255) / 256), 256, 0, stream>>>(rst, res, L[l].b,
                                                                          hnext, N, D);
    hin = hnext;
  }

  float* out_f = (float*)d_out;
  fill_kernel<<<1, 32, 0, stream>>>(out_f, NEG_INF, 1);
  mlp_max_kernel<<<(N + 255) / 256, 256, 0, stream>>>(rst,
      (const float*)d_in[23], (const float*)d_in[24], (const float*)d_in[25],
      (const float*)d_in[26], (const float*)d_in[27], (const float*)d_in[28],
      (const float*)d_in[29], (const float*)d_in[30], out_f, N);
}